// TriPreModel_584115552928
// MI455X (gfx1250) — compile-verified
//
#include <hip/hip_runtime.h>
#include <math.h>

// ---------------------------------------------------------------------------
// TriPreModel GCN stack for gfx1250 (MI455X).
// Dense transforms use V_WMMA_F32_16X16X4_F32 (full fp32 precision).
// Edge aggregation uses global f32 atomics (memory-bound part of the model).
// ---------------------------------------------------------------------------

typedef __attribute__((ext_vector_type(2))) float v2f;
typedef __attribute__((ext_vector_type(8))) float v8f;

#define N_NODES 100000
#define F_IN    128
#define C_ST    64
#define C_TS    64
#define C_1     64
#define C_H1    192   // C_ST + C_TS + C_1
#define C_MID   128
#define N_CLS   16

// ---------------------------------------------------------------------------
// utility: zero fill
// ---------------------------------------------------------------------------
__global__ __launch_bounds__(256) void zero_f32_kernel(float* __restrict__ p, long long n) {
    long long i = (long long)blockIdx.x * blockDim.x + threadIdx.x;
    long long stride = (long long)gridDim.x * blockDim.x;
    for (; i < n; i += stride) p[i] = 0.0f;
}

// ---------------------------------------------------------------------------
// concat [W_st | W_ts | W1] -> Wcat [128 x 192] row-major
// ---------------------------------------------------------------------------
__global__ __launch_bounds__(256) void concat_w_kernel(const float* __restrict__ Wst,
                                                       const float* __restrict__ Wts,
                                                       const float* __restrict__ W1,
                                                       float* __restrict__ Wcat) {
    int i = blockIdx.x * blockDim.x + threadIdx.x;          // over 128*192
    if (i >= F_IN * C_H1) return;
    int k = i / C_H1;
    int c = i - k * C_H1;
    float v;
    if (c < 64)       v = Wst[k * 64 + c];
    else if (c < 128) v = Wts[k * 64 + (c - 64)];
    else              v = W1 [k * 64 + (c - 128)];
    Wcat[i] = v;
}

// ---------------------------------------------------------------------------
// degree counting (f32 atomics into pre-zeroed count buffers)
// ---------------------------------------------------------------------------
__global__ __launch_bounds__(256) void deg_count_kernel(const int* __restrict__ dst,
                                                        const int* __restrict__ rev,
                                                        float* __restrict__ c_st,
                                                        float* __restrict__ c_ts,
                                                        float* __restrict__ c_all,
                                                        int nE) {
    int e = blockIdx.x * blockDim.x + threadIdx.x;
    if (e >= nE) return;
    int d = dst[e];
    if (rev[e]) atomicAdd(&c_ts[d], 1.0f);
    else        atomicAdd(&c_st[d], 1.0f);
    atomicAdd(&c_all[d], 1.0f);
}

// counts -> dinv = rsqrt(count + 1)   (three contiguous arrays, 3N elements)
__global__ __launch_bounds__(256) void deg_finalize_kernel(float* __restrict__ p, int n) {
    int i = blockIdx.x * blockDim.x + threadIdx.x;
    if (i >= n) return;
    p[i] = rsqrtf(p[i] + 1.0f);
}

// ---------------------------------------------------------------------------
// WMMA fp32 GEMM:  out[nRows x (NT*16)] = A[nRows x K] @ W[K x (NT*16)]
// One wave computes a 16-row strip across all NT column tiles.
// A-fragment (16x4 f32): lane<16 -> {A[m][k],A[m][k+1]}, lane>=16 -> {k+2,k+3}
// B-fragment (4x16 f32): lane<16 -> {W[k][n],W[k+1][n]}, lane>=16 -> {k+2,k+3}
// D (16x16 f32): VGPR v, lanes 0-15 -> M=v, lanes 16-31 -> M=v+8.
// ---------------------------------------------------------------------------
template<int NT>
__global__ __launch_bounds__(256) void gemm_wmma_kernel(const float* __restrict__ A,
                                                        const float* __restrict__ W,
                                                        float* __restrict__ out,
                                                        int nRows, int K) {
    const int Co = NT * 16;
    int wave = blockIdx.x * (blockDim.x >> 5) + (threadIdx.x >> 5);
    int lane = threadIdx.x & 31;
    int rowTile = wave;
    if (rowTile * 16 >= nRows) return;        // wave-uniform exit, EXEC stays full

    int m  = lane & 15;
    int hi = lane >> 4;                       // 0 for lanes 0-15, 1 for 16-31

    const float* Arow = A + (size_t)(rowTile * 16 + m) * K;

    v8f acc[NT];
#pragma unroll
    for (int t = 0; t < NT; ++t)
#pragma unroll
        for (int j = 0; j < 8; ++j) acc[t][j] = 0.0f;

    for (int k = 0; k < K; k += 4) {
        int kc = k + 2 * hi;
        v2f a;
        a.x = Arow[kc];
        a.y = Arow[kc + 1];
        const float* Wp = W + (size_t)kc * Co + m;
#pragma unroll
        for (int t = 0; t < NT; ++t) {
            v2f b;
            b.x = Wp[t * 16];
            b.y = Wp[t * 16 + Co];
            acc[t] = __builtin_amdgcn_wmma_f32_16x16x4_f32(
                false, a, false, b, (short)0, acc[t], false, false);
        }
    }

    int rowBase = rowTile * 16 + hi * 8;
#pragma unroll
    for (int t = 0; t < NT; ++t) {
        float* op = out + (size_t)rowBase * Co + t * 16 + m;
#pragma unroll
        for (int v = 0; v < 8; ++v) op[(size_t)v * Co] = acc[t][v];
    }
}

// ---------------------------------------------------------------------------
// layer-1 aggregation: 64 threads per edge.
// Every edge is exactly one of {st, ts} plus the "all" branch:
//   cols [base..base+63] with branch coef, cols [128..191] with all coef.
// ---------------------------------------------------------------------------
__global__ __launch_bounds__(256) void aggregate1_kernel(const float* __restrict__ h1lin,
                                                         float* __restrict__ agg,
                                                         const int* __restrict__ src,
                                                         const int* __restrict__ dst,
                                                         const int* __restrict__ rev,
                                                         const float* __restrict__ d_st,
                                                         const float* __restrict__ d_ts,
                                                         const float* __restrict__ d_all,
                                                         int nE) {
    long long tid = (long long)blockIdx.x * blockDim.x + threadIdx.x;
    long long e = tid >> 6;
    int f = (int)(tid & 63);
    if (e >= nE) return;
    int s = src[e], d = dst[e];
    int r = rev[e];
    float cb = r ? (d_ts[s] * d_ts[d]) : (d_st[s] * d_st[d]);
    float ca = d_all[s] * d_all[d];
    int base = r ? 64 : 0;
    const float* hs = h1lin + (size_t)s * C_H1;
    float* ad = agg + (size_t)d * C_H1;
    atomicAdd(&ad[base + f], hs[base + f] * cb);
    atomicAdd(&ad[128 + f],  hs[128 + f]  * ca);
}

// generic "all-edges" aggregation, F = 1<<shiftF threads per edge
__global__ __launch_bounds__(256) void aggregate_all_kernel(const float* __restrict__ h,
                                                            float* __restrict__ agg,
                                                            const int* __restrict__ src,
                                                            const int* __restrict__ dst,
                                                            const float* __restrict__ d_all,
                                                            int nE, int shiftF) {
    long long tid = (long long)blockIdx.x * blockDim.x + threadIdx.x;
    long long e = tid >> shiftF;
    int F = 1 << shiftF;
    int f = (int)(tid & (F - 1));
    if (e >= nE) return;
    int s = src[e], d = dst[e];
    float ca = d_all[s] * d_all[d];
    atomicAdd(&agg[(size_t)d * F + f], h[(size_t)s * F + f] * ca);
}

// ---------------------------------------------------------------------------
// finalize layer 1: h1 = relu(agg + h1lin * dinv^2 + bias), in-place into agg
// ---------------------------------------------------------------------------
__global__ __launch_bounds__(256) void finalize1_kernel(const float* __restrict__ h1lin,
                                                        float* __restrict__ agg,
                                                        const float* __restrict__ d_st,
                                                        const float* __restrict__ d_ts,
                                                        const float* __restrict__ d_all,
                                                        const float* __restrict__ b_st,
                                                        const float* __restrict__ b_ts,
                                                        const float* __restrict__ b1,
                                                        int n) {
    int i = blockIdx.x * blockDim.x + threadIdx.x;   // over N*192 (< 2^31)
    if (i >= n) return;
    int node = i / C_H1;
    int c = i - node * C_H1;
    float di, bias;
    if (c < 64)       { di = d_st[node];  bias = b_st[c]; }
    else if (c < 128) { di = d_ts[node];  bias = b_ts[c - 64]; }
    else              { di = d_all[node]; bias = b1[c - 128]; }
    float v = agg[i] + h1lin[i] * di * di + bias;
    agg[i] = v > 0.0f ? v : 0.0f;
}

// finalize layer 2 (no relu): h2 = agg + h2lin * dinv_all^2 + b2, in-place
__global__ __launch_bounds__(256) void finalize2_kernel(const float* __restrict__ h2lin,
                                                        float* __restrict__ agg,
                                                        const float* __restrict__ d_all,
                                                        const float* __restrict__ b2,
                                                        int n) {
    int i = blockIdx.x * blockDim.x + threadIdx.x;   // over N*128
    if (i >= n) return;
    int node = i >> 7;
    int c = i & 127;
    float di = d_all[node];
    agg[i] = agg[i] + h2lin[i] * di * di + b2[c];
}

// finalize layer 3 + log_softmax over 16 classes, one thread per node
__global__ __launch_bounds__(256) void finalize3_lsm_kernel(const float* __restrict__ h3lin,
                                                            const float* __restrict__ agg3,
                                                            const float* __restrict__ d_all,
                                                            const float* __restrict__ b3,
                                                            float* __restrict__ out,
                                                            int n) {
    int i = blockIdx.x * blockDim.x + threadIdx.x;
    if (i >= n) return;
    float di = d_all[i];
    float sc = di * di;
    float v[N_CLS];
    float mx = -3.4e38f;
#pragma unroll
    for (int c = 0; c < N_CLS; ++c) {
        float t = agg3[(size_t)i * N_CLS + c] + h3lin[(size_t)i * N_CLS + c] * sc + b3[c];
        v[c] = t;
        mx = t > mx ? t : mx;
    }
    float s = 0.0f;
#pragma unroll
    for (int c = 0; c < N_CLS; ++c) s += __expf(v[c] - mx);
    float l = __logf(s);
#pragma unroll
    for (int c = 0; c < N_CLS; ++c) out[(size_t)i * N_CLS + c] = v[c] - mx - l;
}

// ---------------------------------------------------------------------------
// host-side orchestration
// ---------------------------------------------------------------------------
static inline int cdiv(long long a, long long b) { return (int)((a + b - 1) / b); }

extern "C" void kernel_launch(void* const* d_in, const int* in_sizes, int n_in,
                              void* d_out, int out_size, void* d_ws, size_t ws_size,
                              hipStream_t stream) {
    const float* x     = (const float*)d_in[0];
    const int*   eidx  = (const int*)d_in[1];
    const int*   rev   = (const int*)d_in[2];
    const float* W_st  = (const float*)d_in[3];
    const float* b_st  = (const float*)d_in[4];
    const float* W_ts  = (const float*)d_in[5];
    const float* b_ts  = (const float*)d_in[6];
    const float* W1    = (const float*)d_in[7];
    const float* b1    = (const float*)d_in[8];
    const float* W2    = (const float*)d_in[9];
    const float* b2    = (const float*)d_in[10];
    const float* W3    = (const float*)d_in[11];
    const float* b3    = (const float*)d_in[12];
    float* out = (float*)d_out;

    const int N = in_sizes[0] / F_IN;     // 100000
    const int E = in_sizes[2];            // 1600000
    const int* src = eidx;
    const int* dst = eidx + E;

    // workspace layout (floats)
    float* ws = (float*)d_ws;
    float* dinv_st  = ws;                        // N
    float* dinv_ts  = dinv_st + N;               // N
    float* dinv_all = dinv_ts + N;               // N
    float* Wcat     = dinv_all + N;              // 128*192
    float* P        = Wcat + F_IN * C_H1;        // N*192  (h1lin, later h2lin/h3lin)
    float* Q        = P + (size_t)N * C_H1;      // N*192  (agg1->h1, later agg2->h2)

    const int TPB = 256;

    // ---- degrees ----
    zero_f32_kernel<<<cdiv(3LL * N, TPB * 8), TPB, 0, stream>>>(dinv_st, 3LL * N);
    deg_count_kernel<<<cdiv(E, TPB), TPB, 0, stream>>>(dst, rev, dinv_st, dinv_ts, dinv_all, E);
    deg_finalize_kernel<<<cdiv(3LL * N, TPB), TPB, 0, stream>>>(dinv_st, 3 * N);

    // ---- layer 1 transform: h1lin = x @ [W_st|W_ts|W1] ----
    concat_w_kernel<<<cdiv(F_IN * C_H1, TPB), TPB, 0, stream>>>(W_st, W_ts, W1, Wcat);
    {
        int tiles = cdiv(N, 16);
        gemm_wmma_kernel<12><<<cdiv(tiles, 8), TPB, 0, stream>>>(x, Wcat, P, N, F_IN);
    }

    // ---- layer 1 aggregation + finalize (h1 lives in Q) ----
    zero_f32_kernel<<<cdiv((long long)N * C_H1, TPB * 8), TPB, 0, stream>>>(Q, (long long)N * C_H1);
    aggregate1_kernel<<<cdiv((long long)E * 64, TPB), TPB, 0, stream>>>(
        P, Q, src, dst, rev, dinv_st, dinv_ts, dinv_all, E);
    finalize1_kernel<<<cdiv((long long)N * C_H1, TPB), TPB, 0, stream>>>(
        P, Q, dinv_st, dinv_ts, dinv_all, b_st, b_ts, b1, N * C_H1);

    // ---- layer 2: h2lin = h1 @ W2 -> P, aggregate -> Q, finalize (h2 in Q) ----
    {
        int tiles = cdiv(N, 16);
        gemm_wmma_kernel<8><<<cdiv(tiles, 8), TPB, 0, stream>>>(Q, W2, P, N, C_H1);
    }
    zero_f32_kernel<<<cdiv((long long)N * C_MID, TPB * 8), TPB, 0, stream>>>(Q, (long long)N * C_MID);
    aggregate_all_kernel<<<cdiv((long long)E * C_MID, TPB), TPB, 0, stream>>>(
        P, Q, src, dst, dinv_all, E, 7);
    finalize2_kernel<<<cdiv((long long)N * C_MID, TPB), TPB, 0, stream>>>(
        P, Q, dinv_all, b2, N * C_MID);

    // ---- layer 3: h3lin = h2 @ W3 -> P[0..16N), agg3 -> P[16N..32N) ----
    float* h3lin = P;
    float* agg3  = P + (size_t)N * N_CLS;
    {
        int tiles = cdiv(N, 16);
        gemm_wmma_kernel<1><<<cdiv(tiles, 8), TPB, 0, stream>>>(Q, W3, h3lin, N, C_MID);
    }
    zero_f32_kernel<<<cdiv((long long)N * N_CLS, TPB * 8), TPB, 0, stream>>>(agg3, (long long)N * N_CLS);
    aggregate_all_kernel<<<cdiv((long long)E * N_CLS, TPB), TPB, 0, stream>>>(
        h3lin, agg3, src, dst, dinv_all, E, 4);

    // ---- finalize + log_softmax ----
    finalize3_lsm_kernel<<<cdiv(N, TPB), TPB, 0, stream>>>(h3lin, agg3, dinv_all, b3, out, N);
}